// MyModel_3736621547741
// MI455X (gfx1250) — compile-verified
//
#include <hip/hip_runtime.h>

#define EPSK   1e-6f
#define S_DIM  32
#define D_DIM  128
#define NEG_INF -3.4e38f
#define UPAD   4112   // padded stride for u/w vectors (>= N+1)
#define POBJ_STRIDE 520

typedef float v2f __attribute__((ext_vector_type(2)));
typedef float v8f __attribute__((ext_vector_type(8)));

__device__ __forceinline__ float waveSum(float v) {
#pragma unroll
  for (int m = 16; m > 0; m >>= 1) v += __shfl_xor(v, m, 32);
  return v;
}

__device__ __forceinline__ v8f wmma_f32_16x16x4(v2f a, v2f b, v8f c) {
  // (neg_a, A, neg_b, B, c_mod, C, reuse_a, reuse_b)
  return __builtin_amdgcn_wmma_f32_16x16x4_f32(false, a, false, b, (short)0, c,
                                               false, false);
}

// ---------------------------------------------------------------------------
// 1) Deterministic treated/control split (stable order via block prefix scan)
// ---------------------------------------------------------------------------
__global__ __launch_bounds__(1024) void k_split(const int* __restrict__ t, int N,
                                                int* __restrict__ it_idx,
                                                int* __restrict__ ic_idx,
                                                int* __restrict__ counts,
                                                float* __restrict__ pmeta) {
  __shared__ int sc[1024];
  int tid = threadIdx.x;
  int base = tid * 4;
  int loc[4];
  int c = 0;
#pragma unroll
  for (int e = 0; e < 4; ++e) {
    int idx = base + e;
    int v = (idx < N && t[idx] > 0) ? 1 : 0;
    loc[e] = v;
    c += v;
  }
  sc[tid] = c;
  __syncthreads();
  for (int off = 1; off < 1024; off <<= 1) {
    int v = (tid >= off) ? sc[tid - off] : 0;
    __syncthreads();
    sc[tid] += v;
    __syncthreads();
  }
  int tpre = sc[tid] - c;   // treated strictly before this thread's chunk
  int cpre = base - tpre;   // controls strictly before
#pragma unroll
  for (int e = 0; e < 4; ++e) {
    int idx = base + e;
    if (idx < N) {
      if (loc[e]) it_idx[tpre++] = idx;
      else        ic_idx[cpre++] = idx;
    }
  }
  if (tid == 0) {
    int nt = sc[1023];
    int nc = N - nt;
    counts[0] = nt;
    counts[1] = nc;
    counts[2] = nt * nc;
    pmeta[0] = (float)nt / (float)N;
  }
}

// ---------------------------------------------------------------------------
// 2) Row norms: norms[s*N + n] = sum_d X[n,s,d]^2   (wave per (n,s) row)
// ---------------------------------------------------------------------------
__global__ __launch_bounds__(256) void k_norms(const float* __restrict__ X,
                                               float* __restrict__ norms, int N) {
  int ns = blockIdx.x * 8 + (threadIdx.x >> 5);
  int lane = threadIdx.x & 31;
  if (ns >= N * S_DIM) return;
  const float* row = X + (size_t)ns * D_DIM;
  float4 v = *(const float4*)(row + lane * 4);
  float a = v.x * v.x + v.y * v.y + v.z * v.z + v.w * v.w;
  a = waveSum(a);
  if (lane == 0) {
    int n = ns >> 5;       // S_DIM == 32
    int s = ns & 31;
    norms[(size_t)s * N + n] = a;
  }
}

// ---------------------------------------------------------------------------
// 3) Distance-matrix GEMM via V_WMMA_F32_16X16X4_F32.
//    Block = 128 threads (4 waves), 64x64 output tile; wave owns 32x32 (2x2
//    WMMA tiles). LDS staged k-major so fragments load per the ISA layout:
//      A 16x4 : lanes 0-15 {K=k,k+1}, lanes 16-31 {K=k+2,k+3}
//      B 4x16 : same K striping with N on lanes
//    Also emits per-tile sum/max partials for deterministic stats reduction.
// ---------------------------------------------------------------------------
__global__ __launch_bounds__(128) void k_gemm(
    const float* __restrict__ X, const int* __restrict__ itx,
    const int* __restrict__ icx, const int* __restrict__ counts,
    const float* __restrict__ norms, float* __restrict__ Mbuf,
    float* __restrict__ psum, float* __restrict__ pmax, int N) {
  int nt = counts[0], nc = counts[1], ntc = counts[2];
  int s = blockIdx.z;
  int bm = blockIdx.y * 64, bn = blockIdx.x * 64;
  int tileId = blockIdx.y * gridDim.x + blockIdx.x;   // < 4096
  if (bm >= nt || bn >= nc) {
    if (threadIdx.x == 0) {
      psum[s * 4096 + tileId] = 0.0f;
      pmax[s * 4096 + tileId] = NEG_INF;
    }
    return;
  }
  __shared__ float As[32 * 64];   // [k][row]
  __shared__ float Bs[32 * 64];   // [k][col]
  __shared__ float nTs[64], nCs[64];
  __shared__ int rIdx[64], cIdx[64];
  __shared__ float red[128];

  int tid = threadIdx.x;
  int lane = tid & 31;
  int wvid = tid >> 5;                  // 0..3
  int wm = (wvid >> 1) * 32;            // wave M offset in 64x64 tile
  int wn = (wvid & 1) * 32;             // wave N offset

  if (tid < 64) {
    int i = bm + tid;
    int gr = itx[i < nt ? i : (nt - 1)];
    rIdx[tid] = gr;
    nTs[tid] = norms[(size_t)s * N + gr];
  } else {
    int j = bn + (tid - 64);
    int gc = icx[j < nc ? j : (nc - 1)];
    cIdx[tid - 64] = gc;
    nCs[tid - 64] = norms[(size_t)s * N + gc];
  }

  v8f accz = {0.f, 0.f, 0.f, 0.f, 0.f, 0.f, 0.f, 0.f};
  v8f acc00 = accz, acc01 = accz, acc10 = accz, acc11 = accz;

  int r = tid & 63;       // row/col within tile this thread stages
  int q = tid >> 6;       // which 16-wide k half
  int l15 = lane & 15;
  int hi = (lane >> 4) & 1;
  int koff2 = hi * 2;

  for (int k0 = 0; k0 < D_DIM; k0 += 32) {
    __syncthreads();      // previous compute done; rIdx/cIdx ready (1st iter)
    const float* Arow = X + ((size_t)rIdx[r] * S_DIM + s) * D_DIM + k0;
    const float* Brow = X + ((size_t)cIdx[r] * S_DIM + s) * D_DIM + k0;
#pragma unroll
    for (int cb = 0; cb < 4; ++cb) {
      int kk = q * 16 + cb * 4;
      float4 av = *(const float4*)(Arow + kk);
      As[(kk + 0) * 64 + r] = av.x;
      As[(kk + 1) * 64 + r] = av.y;
      As[(kk + 2) * 64 + r] = av.z;
      As[(kk + 3) * 64 + r] = av.w;
      float4 bv = *(const float4*)(Brow + kk);
      Bs[(kk + 0) * 64 + r] = bv.x;
      Bs[(kk + 1) * 64 + r] = bv.y;
      Bs[(kk + 2) * 64 + r] = bv.z;
      Bs[(kk + 3) * 64 + r] = bv.w;
    }
    __syncthreads();
#pragma unroll
    for (int kk = 0; kk < 32; kk += 4) {
      int kb = (kk + koff2) * 64;
      v2f a0, a1, b0, b1;
      a0.x = As[kb + wm + l15];        a0.y = As[kb + 64 + wm + l15];
      a1.x = As[kb + wm + 16 + l15];   a1.y = As[kb + 64 + wm + 16 + l15];
      b0.x = Bs[kb + wn + l15];        b0.y = Bs[kb + 64 + wn + l15];
      b1.x = Bs[kb + wn + 16 + l15];   b1.y = Bs[kb + 64 + wn + 16 + l15];
      acc00 = wmma_f32_16x16x4(a0, b0, acc00);
      acc01 = wmma_f32_16x16x4(a0, b1, acc01);
      acc10 = wmma_f32_16x16x4(a1, b0, acc10);
      acc11 = wmma_f32_16x16x4(a1, b1, acc11);
    }
  }

  // Epilogue: M = ||x||^2 + ||y||^2 - 2*dot ; write + local sum/max
  float lsum = 0.0f, lmax = NEG_INF;
#define EPI(ACC, TI, TJ)                                                     \
  _Pragma("unroll") for (int rr = 0; rr < 8; ++rr) {                         \
    int m = wm + (TI)*16 + rr + hi * 8;                                      \
    int n = wn + (TJ)*16 + l15;                                              \
    int gi = bm + m, gj = bn + n;                                            \
    if (gi < nt && gj < nc) {                                                \
      float Mij = nTs[m] + nCs[n] - 2.0f * (ACC)[rr];                        \
      Mbuf[(size_t)s * (size_t)ntc + (size_t)gi * (size_t)nc + gj] = Mij;    \
      lsum += Mij;                                                           \
      lmax = fmaxf(lmax, Mij);                                               \
    }                                                                        \
  }
  EPI(acc00, 0, 0)
  EPI(acc01, 0, 1)
  EPI(acc10, 1, 0)
  EPI(acc11, 1, 1)
#undef EPI

  red[tid] = lsum;
  __syncthreads();
  for (int o2 = 64; o2 > 0; o2 >>= 1) {
    if (tid < o2) red[tid] += red[tid + o2];
    __syncthreads();
  }
  if (tid == 0) psum[s * 4096 + tileId] = red[0];
  __syncthreads();
  red[tid] = lmax;
  __syncthreads();
  for (int o2 = 64; o2 > 0; o2 >>= 1) {
    if (tid < o2) red[tid] = fmaxf(red[tid], red[tid + o2]);
    __syncthreads();
  }
  if (tid == 0) pmax[s * 4096 + tileId] = red[0];
}

// ---------------------------------------------------------------------------
// 4) Per-timestep stats: delta = max(M), eff_lam = nt*nc/sum(M), K_delta.
//    Also initialize u = a.
// ---------------------------------------------------------------------------
__global__ __launch_bounds__(256) void k_stats(const int* __restrict__ counts,
                                               const float* __restrict__ pmeta,
                                               const float* __restrict__ psum,
                                               const float* __restrict__ pmax,
                                               float* __restrict__ scal,
                                               float* __restrict__ uvec) {
  int s = blockIdx.x;
  int tid = threadIdx.x;
  __shared__ float rs[256], rm[256];
  float a = 0.0f, m = NEG_INF;
  for (int k = tid; k < 4096; k += 256) {
    a += psum[s * 4096 + k];
    m = fmaxf(m, pmax[s * 4096 + k]);
  }
  rs[tid] = a;
  rm[tid] = m;
  __syncthreads();
  for (int o2 = 128; o2 > 0; o2 >>= 1) {
    if (tid < o2) {
      rs[tid] += rs[tid + o2];
      rm[tid] = fmaxf(rm[tid], rm[tid + o2]);
    }
    __syncthreads();
  }
  int nt = counts[0], nc = counts[1];
  float p = pmeta[0];
  if (tid == 0) {
    float Msum = rs[0];
    float delta = rm[0];
    float lam = ((float)nt * (float)nc) / Msum;    // LAM=1 / mean(M)
    scal[s * 4 + 0] = delta;
    scal[s * 4 + 1] = lam;
    scal[s * 4 + 2] = __expf(-lam * delta) + EPSK; // K_delta
  }
  float* u = uvec + (size_t)s * UPAD;
  float ai = p / (float)nt;
  for (int i = tid; i <= nt; i += 256) u[i] = (i < nt) ? ai : (1.0f - p);
}

// ---------------------------------------------------------------------------
// 5) Half-iteration 1: w = b / (K^T u). Thread-per-column (coalesced sweep
//    down M's rows), u staged through LDS in 256-chunks. exp on the fly.
// ---------------------------------------------------------------------------
__global__ __launch_bounds__(256) void k_half1(const float* __restrict__ Mbuf,
                                               const int* __restrict__ counts,
                                               const float* __restrict__ pmeta,
                                               const float* __restrict__ scal,
                                               const float* __restrict__ uvec,
                                               float* __restrict__ wvec) {
  int nt = counts[0], nc = counts[1], ntc = counts[2];
  float p = pmeta[0];
  int s = blockIdx.y;
  int tid = threadIdx.x;
  int j = blockIdx.x * 256 + tid;
  float lam = scal[s * 4 + 1];
  float Kd = scal[s * 4 + 2];
  const float* u = uvec + (size_t)s * UPAD;
  const float* Mcol = Mbuf + (size_t)s * (size_t)ntc;
  __shared__ float uls[256];
  float acc = 0.0f;
  for (int i0 = 0; i0 < nt; i0 += 256) {
    int ii = i0 + tid;
    uls[tid] = (ii < nt) ? u[ii] : 0.0f;
    __syncthreads();
    int lim = min(256, nt - i0);
    if (j < nc) {
      const float* mp = Mcol + (size_t)i0 * (size_t)nc + j;
      for (int k = 0; k < lim; ++k)
        acc += (__expf(-lam * mp[(size_t)k * (size_t)nc]) + EPSK) * uls[k];
    } else if (j == nc) {
      for (int k = 0; k < lim; ++k) acc += uls[k];
    }
    __syncthreads();
  }
  float unt = u[nt];
  float* w = wvec + (size_t)s * UPAD;
  if (j < nc) {
    w[j] = ((1.0f - p) / (float)nc) / (acc + Kd * unt);
  } else if (j == nc) {
    w[j] = p / (Kd * acc + (1.0f + EPSK) * unt);
  }
}

// ---------------------------------------------------------------------------
// 6) Half-iteration 2: u = a / (K w). Wave-per-row, lane-strided columns.
// ---------------------------------------------------------------------------
__global__ __launch_bounds__(256) void k_half2(const float* __restrict__ Mbuf,
                                               const int* __restrict__ counts,
                                               const float* __restrict__ pmeta,
                                               const float* __restrict__ scal,
                                               float* __restrict__ uvec,
                                               const float* __restrict__ wvec) {
  int nt = counts[0], nc = counts[1], ntc = counts[2];
  float p = pmeta[0];
  int s = blockIdx.y;
  int lane = threadIdx.x & 31;
  int wvid = threadIdx.x >> 5;
  int row = blockIdx.x * 8 + wvid;
  if (row > nt) return;
  float lam = scal[s * 4 + 1];
  float Kd = scal[s * 4 + 2];
  const float* w = wvec + (size_t)s * UPAD;
  float* u = uvec + (size_t)s * UPAD;
  float acc = 0.0f;
  if (row < nt) {
    const float* Mr = Mbuf + (size_t)s * (size_t)ntc + (size_t)row * (size_t)nc;
    for (int jj = lane; jj < nc; jj += 32)
      acc += (__expf(-lam * Mr[jj]) + EPSK) * w[jj];
    acc = waveSum(acc);
    if (lane == 0) u[row] = (p / (float)nt) / (acc + Kd * w[nc]);
  } else {  // augmented row
    for (int jj = lane; jj < nc; jj += 32) acc += w[jj];
    acc = waveSum(acc);
    if (lane == 0) u[nt] = (1.0f - p) / (Kd * acc + (1.0f + EPSK) * w[nc]);
  }
}

// ---------------------------------------------------------------------------
// 7) Objective core partials: sum_ij u_i K_ij M_ij v_j  (augmented terms in
//    the final kernel). Wave-per-row; deterministic per-block partials.
// ---------------------------------------------------------------------------
__global__ __launch_bounds__(256) void k_obj(const float* __restrict__ Mbuf,
                                             const int* __restrict__ counts,
                                             const float* __restrict__ scal,
                                             const float* __restrict__ uvec,
                                             const float* __restrict__ wvec,
                                             float* __restrict__ pobj) {
  int nt = counts[0], nc = counts[1], ntc = counts[2];
  int s = blockIdx.y;
  int lane = threadIdx.x & 31;
  int wvid = threadIdx.x >> 5;
  int row = blockIdx.x * 8 + wvid;
  float lam = scal[s * 4 + 1];
  const float* w = wvec + (size_t)s * UPAD;
  const float* u = uvec + (size_t)s * UPAD;
  __shared__ float part[8];
  float contrib = 0.0f;
  if (row < nt) {
    const float* Mr = Mbuf + (size_t)s * (size_t)ntc + (size_t)row * (size_t)nc;
    float rs = 0.0f;
    for (int jj = lane; jj < nc; jj += 32) {
      float m = Mr[jj];
      rs += (__expf(-lam * m) + EPSK) * m * w[jj];
    }
    rs = waveSum(rs);
    contrib = u[row] * rs;
  }
  if (lane == 0) part[wvid] = contrib;
  __syncthreads();
  if (threadIdx.x == 0) {
    float b = 0.0f;
    for (int k = 0; k < 8; ++k) b += part[k];
    pobj[s * POBJ_STRIDE + blockIdx.x] = b;
  }
}

// ---------------------------------------------------------------------------
// 8) Final deterministic reduce: add augmented-row/col terms, x2, sum over s.
// ---------------------------------------------------------------------------
__global__ __launch_bounds__(256) void k_final(const int* __restrict__ counts,
                                               const float* __restrict__ scal,
                                               const float* __restrict__ uvec,
                                               const float* __restrict__ wvec,
                                               const float* __restrict__ pobj,
                                               float* __restrict__ out) {
  __shared__ float os[32];
  int tid = threadIdx.x;
  int nt = counts[0], nc = counts[1];
  if (tid < 32) {
    int s = tid;
    float delta = scal[s * 4 + 0];
    float Kd = scal[s * 4 + 2];
    const float* u = uvec + (size_t)s * UPAD;
    const float* w = wvec + (size_t)s * UPAD;
    float core = 0.0f;
    for (int k = 0; k < 513; ++k) core += pobj[s * POBJ_STRIDE + k];
    float Us = 0.0f;
    for (int i = 0; i < nt; ++i) Us += u[i];
    float Ws = 0.0f;
    for (int j = 0; j < nc; ++j) Ws += w[j];
    os[s] = core + u[nt] * Kd * delta * Ws + w[nc] * Kd * delta * Us;
  }
  __syncthreads();
  if (tid == 0) {
    float tot = 0.0f;
    for (int s = 0; s < 32; ++s) tot += os[s];
    out[0] = 2.0f * tot;
  }
}

// ---------------------------------------------------------------------------
extern "C" void kernel_launch(void* const* d_in, const int* in_sizes, int n_in,
                              void* d_out, int out_size, void* d_ws,
                              size_t ws_size, hipStream_t stream) {
  (void)n_in; (void)out_size; (void)ws_size;
  const float* X = (const float*)d_in[0];
  const int* t = (const int*)d_in[1];
  int N = in_sizes[1];  // 4096

  char* base = (char*)d_ws;
  size_t o = 0;
  auto take = [&](size_t bytes) -> void* {
    void* p = base + o;
    o = (o + bytes + 255) & ~(size_t)255;
    return p;
  };
  int* it_idx   = (int*)take((size_t)N * 4);
  int* ic_idx   = (int*)take((size_t)N * 4);
  int* counts   = (int*)take(16);
  float* pmeta  = (float*)take(16);
  float* norms  = (float*)take((size_t)N * S_DIM * 4);
  float* scal   = (float*)take(32 * 4 * 4);
  float* psum   = (float*)take((size_t)32 * 4096 * 4);
  float* pmax   = (float*)take((size_t)32 * 4096 * 4);
  float* uvec   = (float*)take((size_t)32 * UPAD * 4);
  float* wvec   = (float*)take((size_t)32 * UPAD * 4);
  float* pobj   = (float*)take((size_t)32 * POBJ_STRIDE * 4);
  float* Mbuf   = (float*)take((size_t)32 * (size_t)(N / 2) * (size_t)(N / 2) * 4);

  k_split<<<1, 1024, 0, stream>>>(t, N, it_idx, ic_idx, counts, pmeta);
  k_norms<<<(N * S_DIM) / 8, 256, 0, stream>>>(X, norms, N);

  dim3 gg((N + 63) / 64, (N + 63) / 64, S_DIM);
  k_gemm<<<gg, 128, 0, stream>>>(X, it_idx, ic_idx, counts, norms, Mbuf, psum,
                                 pmax, N);
  k_stats<<<S_DIM, 256, 0, stream>>>(counts, pmeta, psum, pmax, scal, uvec);

  dim3 g1((N + 1 + 255) / 256, S_DIM);  // 17 x 32
  dim3 g2((N + 1 + 7) / 8, S_DIM);      // 513 x 32
  for (int it = 0; it < 20; ++it) {
    k_half1<<<g1, 256, 0, stream>>>(Mbuf, counts, pmeta, scal, uvec, wvec);
    k_half2<<<g2, 256, 0, stream>>>(Mbuf, counts, pmeta, scal, uvec, wvec);
  }
  k_half1<<<g1, 256, 0, stream>>>(Mbuf, counts, pmeta, scal, uvec, wvec);  // v
  k_obj<<<g2, 256, 0, stream>>>(Mbuf, counts, scal, uvec, wvec, pobj);
  k_final<<<1, 256, 0, stream>>>(counts, scal, uvec, wvec, pobj,
                                 (float*)d_out);
}